// DecoupledGraphPooling_80135499809390
// MI455X (gfx1250) — compile-verified
//
#include <hip/hip_runtime.h>
#include <hip/hip_bf16.h>

// ---------------------------------------------------------------------------
// Problem constants (fixed by setup_inputs: B=4, N=4096, D=768, k=2048)
// ---------------------------------------------------------------------------
#define BB 4
#define NN 4096
#define DD 768
#define KK 2048

typedef __attribute__((ext_vector_type(2))) float v2f;
typedef __attribute__((ext_vector_type(4))) float v4f;
typedef __attribute__((ext_vector_type(8))) float v8f;
typedef __attribute__((ext_vector_type(4))) int   i4;

// Async global->LDS copy path (gfx1250): guarded so compile can't regress.
#if defined(__has_builtin)
#  if __has_builtin(__builtin_amdgcn_global_load_async_to_lds_b128) && \
      __has_builtin(__builtin_amdgcn_s_wait_asynccnt)
#    define USE_ASYNC_LDS 1
#  endif
#endif
#ifndef USE_ASYNC_LDS
#  define USE_ASYNC_LDS 0
#endif

// Builtin wants AS(1) int4* (global) and AS(3) int4* (LDS).
typedef __attribute__((address_space(1))) i4 gi4;
typedef __attribute__((address_space(3))) i4 li4;

// ---------------------------------------------------------------------------
// Kernel 1: scores[b][n] = sigmoid( dot(h[b,n,:], sf[b,:]) )
// One wave per 16-row tile, V_WMMA_F32_16X16X4_F32 accumulation (full f32).
// f32 A-matrix 16x4 layout (ISA 7.12.2): lane L holds row M=L%16,
// K-offset = 2*(L/16).  B (4x16) = sf chunk replicated across all N columns,
// so every accumulator column equals the dot product.
// C/D layout: acc[j], lanes 0-15 -> M=j ; lanes 16-31 -> M=8+j.
// ---------------------------------------------------------------------------
__global__ void __launch_bounds__(32)
scores_wmma_kernel(const float* __restrict__ h,
                   const float* __restrict__ sf,
                   float* __restrict__ scores) {
    const int lane   = threadIdx.x;      // 0..31
    const int lane16 = lane & 15;
    const int half   = lane >> 4;        // 0 or 1
    const int b      = blockIdx.y;
    const int row0   = blockIdx.x * 16;

    const float* __restrict__ hrow = h  + ((size_t)b * NN + row0 + lane16) * DD;
    const float* __restrict__ sfb  = sf + (size_t)b * DD;
    const int kofs = half * 2;           // lanes 16-31 supply K+2, K+3

    v8f acc0 = {}, acc1 = {}, acc2 = {}, acc3 = {};
    for (int k0 = 0; k0 < DD; k0 += 16) {
        v2f a0, a1, a2, a3, b0, b1, b2, b3;
        a0.x = hrow[k0 +  0 + kofs]; a0.y = hrow[k0 +  1 + kofs];
        a1.x = hrow[k0 +  4 + kofs]; a1.y = hrow[k0 +  5 + kofs];
        a2.x = hrow[k0 +  8 + kofs]; a2.y = hrow[k0 +  9 + kofs];
        a3.x = hrow[k0 + 12 + kofs]; a3.y = hrow[k0 + 13 + kofs];
        b0.x = sfb[k0 +  0 + kofs];  b0.y = sfb[k0 +  1 + kofs];
        b1.x = sfb[k0 +  4 + kofs];  b1.y = sfb[k0 +  5 + kofs];
        b2.x = sfb[k0 +  8 + kofs];  b2.y = sfb[k0 +  9 + kofs];
        b3.x = sfb[k0 + 12 + kofs];  b3.y = sfb[k0 + 13 + kofs];
        acc0 = __builtin_amdgcn_wmma_f32_16x16x4_f32(false, a0, false, b0, (short)0, acc0, false, false);
        acc1 = __builtin_amdgcn_wmma_f32_16x16x4_f32(false, a1, false, b1, (short)0, acc1, false, false);
        acc2 = __builtin_amdgcn_wmma_f32_16x16x4_f32(false, a2, false, b2, (short)0, acc2, false, false);
        acc3 = __builtin_amdgcn_wmma_f32_16x16x4_f32(false, a3, false, b3, (short)0, acc3, false, false);
    }
    #pragma unroll
    for (int j = 0; j < 8; ++j)
        acc0[j] = acc0[j] + acc1[j] + acc2[j] + acc3[j];

    if (lane16 == 0) {
        float* srow = scores + (size_t)b * NN + row0 + half * 8;
        #pragma unroll
        for (int j = 0; j < 8; ++j) {
            float w = acc0[j];
            srow[j] = 1.0f / (1.0f + __expf(-w));
        }
    }
}

// ---------------------------------------------------------------------------
// Kernel 2: per-batch full bitonic sort of 4096 (score, index) pairs in LDS,
// comparator: value descending, ties -> lower index (matches lax.top_k).
// One 1024-thread workgroup per batch; emits first KK sorted pairs.
// ---------------------------------------------------------------------------
__global__ void __launch_bounds__(1024)
topk_sort_kernel(const float* __restrict__ scores,
                 float* __restrict__ out_vals,
                 int*   __restrict__ out_idx) {
    __shared__ float sv[NN];
    __shared__ int   si[NN];
    const int b = blockIdx.x;
    const int t = threadIdx.x;

    for (int i = t; i < NN; i += 1024) {
        sv[i] = scores[(size_t)b * NN + i];
        si[i] = i;
    }
    __syncthreads();

    for (int size = 2; size <= NN; size <<= 1) {
        for (int stride = size >> 1; stride > 0; stride >>= 1) {
            for (int base = t; base < NN / 2; base += 1024) {
                const int i = 2 * base - (base & (stride - 1));
                const int j = i + stride;
                const bool dir = ((i & size) == 0);
                float vi = sv[i], vj = sv[j];
                int   ii = si[i], ij = si[j];
                const bool before = (vi > vj) || (vi == vj && ii < ij);
                const bool keep = dir ? before : !before;
                if (!keep) {
                    sv[i] = vj; sv[j] = vi;
                    si[i] = ij; si[j] = ii;
                }
            }
            __syncthreads();
        }
    }
    for (int i = t; i < KK; i += 1024) {
        out_vals[(size_t)b * KK + i] = sv[i];
        out_idx [(size_t)b * KK + i] = si[i];
    }
}

// ---------------------------------------------------------------------------
// Row staging helper: copy one NN-float row from global into LDS.
// Async (GLOBAL_LOAD_ASYNC_TO_LDS_B128, ASYNCcnt-tracked) when available,
// else coalesced b128 load + ds_store.  Caller must wait + __syncthreads().
// ---------------------------------------------------------------------------
__device__ __forceinline__ void stage_row(const float* __restrict__ src,
                                          float* __restrict__ dst_lds,
                                          int tid) {
#if USE_ASYNC_LDS
    for (int j = tid * 4; j < NN; j += 256 * 4) {
        __builtin_amdgcn_global_load_async_to_lds_b128(
            (gi4*)(src + j), (li4*)(dst_lds + j), 0, 0);
    }
#else
    for (int j = tid; j < NN / 4; j += 256) {
        v4f x = __builtin_nontemporal_load((const v4f*)src + j);
        *((v4f*)dst_lds + j) = x;
    }
#endif
}

// ---------------------------------------------------------------------------
// Kernel 3: square row+column gathers of g1, g2.
// out[b][i][j] = g[b][idx[i]][idx[j]].  One block per output row: stage both
// 16 KB source rows into LDS (coalesced 16 B transfers), then the 2048 random
// column gathers hit the 64-bank LDS instead of replaying vector-cache lines.
// Stores are coalesced + non-temporal (outputs are write-once; keep the
// 192 MB L2 for gather working set).
// ---------------------------------------------------------------------------
__global__ void __launch_bounds__(256)
gather_rc_kernel(const float* __restrict__ g1,
                 const float* __restrict__ g2,
                 const int*   __restrict__ idx,
                 float* __restrict__ out1,
                 float* __restrict__ out2) {
    __shared__ int   sidx[KK];
    __shared__ float row1[NN];
    __shared__ float row2[NN];
    const int b = blockIdx.y;
    const int i = blockIdx.x;                 // output row 0..KK-1
    const int t = threadIdx.x;
    const int* __restrict__ idxb = idx + (size_t)b * KK;

    const int ri = idxb[i];
    const size_t srow = ((size_t)b * NN + ri) * NN;

    for (int j = t; j < KK; j += 256) sidx[j] = idxb[j];
    stage_row(g1 + srow, row1, t);
    stage_row(g2 + srow, row2, t);
#if USE_ASYNC_LDS
    __builtin_amdgcn_s_wait_asynccnt(0);
#endif
    __syncthreads();

    const size_t orow = ((size_t)b * KK + i) * KK;
    float* __restrict__ o1 = out1 + orow;
    float* __restrict__ o2 = out2 + orow;
    for (int j = t; j < KK; j += 256) {
        const int c = sidx[j];
        __builtin_nontemporal_store(row1[c], o1 + j);
        __builtin_nontemporal_store(row2[c], o2 + j);
    }
}

// ---------------------------------------------------------------------------
// Kernel 4: contiguous row gathers, b128 + non-temporal both directions.
//   g_mask[b][i][:] = g3[b][idx[i]][:]            (k x N)
//   new_h [b][i][:] = h [b][idx[i]][:] * vals[i]  (k x D)
// ---------------------------------------------------------------------------
__global__ void __launch_bounds__(256)
gather_rows_kernel(const float* __restrict__ g3,
                   const float* __restrict__ h,
                   const int*   __restrict__ idx,
                   const float* __restrict__ vals,
                   float* __restrict__ gmask,
                   float* __restrict__ newh) {
    const int b = blockIdx.y;
    const int i = blockIdx.x;
    const int   ri = idx [(size_t)b * KK + i];
    const float v  = vals[(size_t)b * KK + i];

    const v4f* __restrict__ s3 = (const v4f*)(g3 + ((size_t)b * NN + ri) * NN);
    v4f* __restrict__ d3       = (v4f*)(gmask + ((size_t)b * KK + i) * NN);
    for (int j = threadIdx.x; j < NN / 4; j += 256) {
        v4f x = __builtin_nontemporal_load(s3 + j);
        __builtin_nontemporal_store(x, d3 + j);
    }

    const v4f* __restrict__ sh = (const v4f*)(h + ((size_t)b * NN + ri) * DD);
    v4f* __restrict__ dh       = (v4f*)(newh + ((size_t)b * KK + i) * DD);
    for (int j = threadIdx.x; j < DD / 4; j += 256) {
        v4f x = __builtin_nontemporal_load(sh + j);
        __builtin_nontemporal_store(x * v, dh + j);
    }
}

// ---------------------------------------------------------------------------
// Launch
// ---------------------------------------------------------------------------
extern "C" void kernel_launch(void* const* d_in, const int* in_sizes, int n_in,
                              void* d_out, int out_size, void* d_ws, size_t ws_size,
                              hipStream_t stream) {
    const float* g1 = (const float*)d_in[0];
    const float* g2 = (const float*)d_in[1];
    const float* g3 = (const float*)d_in[2];
    const float* h  = (const float*)d_in[3];
    const float* sf = (const float*)d_in[4];
    // d_in[5] is k == 2048 (compile-time constant KK)

    // Workspace layout
    float* scores = (float*)d_ws;                        // B*N floats
    float* vals   = scores + (size_t)BB * NN;            // B*K floats
    int*   idx    = (int*)(vals + (size_t)BB * KK);      // B*K ints

    // Output layout: [g_section | g_sentence | g_mask | new_h]
    float* out        = (float*)d_out;
    float* g_section  = out;
    float* g_sentence = g_section  + (size_t)BB * KK * KK;
    float* g_mask     = g_sentence + (size_t)BB * KK * KK;
    float* new_h      = g_mask     + (size_t)BB * KK * NN;

    // 1) scores via f32 WMMA (one wave per 16 rows)
    scores_wmma_kernel<<<dim3(NN / 16, BB), dim3(32), 0, stream>>>(h, sf, scores);

    // 2) per-batch top-k via full bitonic sort in LDS
    topk_sort_kernel<<<dim3(BB), dim3(1024), 0, stream>>>(scores, vals, idx);

    // 3) k x k row+column gathers of g1, g2 (LDS-staged rows)
    gather_rc_kernel<<<dim3(KK, BB), dim3(256), 0, stream>>>(g1, g2, idx, g_section, g_sentence);

    // 4) contiguous row gathers (g_mask) + scaled rows (new_h)
    gather_rows_kernel<<<dim3(KK, BB), dim3(256), 0, stream>>>(g3, h, idx, vals, g_mask, new_h);
}